// MultiResHashEncoder_91010357002262
// MI455X (gfx1250) — compile-verified
//
#include <hip/hip_runtime.h>
#include <cmath>
#include <cstdint>

// Multi-resolution hash grid encoder (Instant-NGP style), MI455X / gfx1250.
//
// Perf model: tables (12 x 2^17 x 2 x 4B = 12.6 MB) are L2-resident (192 MB
// L2); the 1.5 GB of corner gathers are L2-bound. The 192 MB output stream
// uses non-temporal stores (TH_STORE_NT) so it does not evict the tables.
// Coordinates are staged memory->LDS with the gfx1250 async engine
// (global_load_async_to_lds_b128, ASYNCcnt); each wave32 stages its own 32
// points so a single s_wait_asynccnt suffices (no barrier), then lanes read
// their point with ds_load_b96 via an explicit addrspace(3) pointer. The 12
// levels are fully unrolled -> deep global_load_b64 gather pipeline.

#define NUM_LEVELS   12
#define LOG2_HASH    17
#define HASHMAP_SIZE (1u << LOG2_HASH)
#define HASH_MASK    (HASHMAP_SIZE - 1u)
#define BLOCK        256

typedef __attribute__((ext_vector_type(2))) float f32x2;
typedef __attribute__((address_space(3))) const float* lds_cfp;

struct LevelRes {
    float r[NUM_LEVELS];   // per-level grid resolution (integral value, stored as f32)
};

__global__ __launch_bounds__(BLOCK)
void MultiResHashEncoder_kernel(const float* __restrict__ x,
                                const f32x2* __restrict__ tables,
                                f32x2* __restrict__ out,
                                int n, LevelRes lr)
{
    __shared__ float smem[BLOCK * 3];   // 3 KB: this block's coordinates

    const int tid         = (int)threadIdx.x;
    const int block_start = (int)blockIdx.x * BLOCK;
    const int i           = block_start + tid;

    float px, py, pz;

    if (block_start + BLOCK <= n) {
        // ---- Async staged path (all points in this block valid) ----
        // Each wave32 stages its own 32 points = 384 B = 24 lanes x 16 B.
        const int      lane     = tid & 31;
        const int      wav      = tid >> 5;
        const unsigned lds_base = (unsigned)(uintptr_t)(&smem[0]); // LDS byte offset
        if (lane < 24) {
            unsigned lds_addr = lds_base + (unsigned)(wav * 384 + lane * 16);
            unsigned long long gaddr =
                (unsigned long long)(uintptr_t)x +
                (unsigned long long)((size_t)block_start * 12u + (size_t)(wav * 384 + lane * 16));
            asm volatile("global_load_async_to_lds_b128 %0, %1, off"
                         :: "v"(lds_addr), "v"(gaddr)
                         : "memory");
        }
        // Producer == consumer wave: waiting on our own ASYNCcnt is enough.
        asm volatile("s_wait_asynccnt 0" ::: "memory");
        // Explicit addrspace(3) read -> ds_load_b96 (keeps the compiler from
        // merging this with the tail path into a generic/flat load).
        lds_cfp sp = (lds_cfp)(lds_base + (unsigned)(12 * tid));
        px = sp[0];
        py = sp[1];
        pz = sp[2];
    } else {
        // ---- Tail block: direct global reads with bounds check ----
        if (i >= n) return;
        px = x[3 * i + 0];
        py = x[3 * i + 1];
        pz = x[3 * i + 2];
    }

    const unsigned P0 = 73856093u, P1 = 19349663u, P2 = 83492791u;

#pragma unroll
    for (int l = 0; l < NUM_LEVELS; ++l) {
        const float r  = lr.r[l];
        // xs = x * res (f32, matches reference); x in [0,1) so xs >= 0.
        const float sx = px * r, sy = py * r, sz = pz * r;
        const float fx = floorf(sx), fy = floorf(sy), fz = floorf(sz);
        const int   ix = (int)fx, iy = (int)fy, iz = (int)fz;
        const float wx = sx - fx, wy = sy - fy, wz = sz - fz;

        // Spatial hash with int32 wraparound. Since 2^17 | 2^32, the low 17
        // bits of the uint32 product/XOR equal the reference's floor-mod of
        // the int32 value exactly. Corner+1 hash = corner hash + prime.
        const unsigned hx0 = (unsigned)ix * P0, hx1 = hx0 + P0;
        const unsigned hy0 = (unsigned)iy * P1, hy1 = hy0 + P1;
        const unsigned hz0 = (unsigned)iz * P2, hz1 = hz0 + P2;

        const f32x2* __restrict__ tbl = tables + (size_t)l * HASHMAP_SIZE;

        // 8 corner gathers (global_load_b64 each); reference OFFSETS order is
        // (bx,by,bz) = 000,001,010,011,100,101,110,111.
        const f32x2 f000 = tbl[(hx0 ^ hy0 ^ hz0) & HASH_MASK];
        const f32x2 f001 = tbl[(hx0 ^ hy0 ^ hz1) & HASH_MASK];
        const f32x2 f010 = tbl[(hx0 ^ hy1 ^ hz0) & HASH_MASK];
        const f32x2 f011 = tbl[(hx0 ^ hy1 ^ hz1) & HASH_MASK];
        const f32x2 f100 = tbl[(hx1 ^ hy0 ^ hz0) & HASH_MASK];
        const f32x2 f101 = tbl[(hx1 ^ hy0 ^ hz1) & HASH_MASK];
        const f32x2 f110 = tbl[(hx1 ^ hy1 ^ hz0) & HASH_MASK];
        const f32x2 f111 = tbl[(hx1 ^ hy1 ^ hz1) & HASH_MASK];

        const float ux = 1.0f - wx, uy = 1.0f - wy, uz = 1.0f - wz;
        const float w000 = ux * uy * uz;
        const float w001 = ux * uy * wz;
        const float w010 = ux * wy * uz;
        const float w011 = ux * wy * wz;
        const float w100 = wx * uy * uz;
        const float w101 = wx * uy * wz;
        const float w110 = wx * wy * uz;
        const float w111 = wx * wy * wz;

        f32x2 acc = f000 * w000;
        acc += f001 * w001;
        acc += f010 * w010;
        acc += f011 * w011;
        acc += f100 * w100;
        acc += f101 * w101;
        acc += f110 * w110;
        acc += f111 * w111;

        // Level-major output [L, N, 2]. Non-temporal (TH_STORE_NT) so the
        // 192 MB output stream does not evict the L2-resident hash tables.
        f32x2* o = out + (size_t)l * (size_t)n + (size_t)i;
        __builtin_nontemporal_store(acc, o);
    }
}

extern "C" void kernel_launch(void* const* d_in, const int* in_sizes, int n_in,
                              void* d_out, int out_size, void* d_ws, size_t ws_size,
                              hipStream_t stream)
{
    (void)n_in; (void)out_size; (void)d_ws; (void)ws_size;

    const float* x      = (const float*)d_in[0];   // [N,3] f32
    const f32x2* tables = (const f32x2*)d_in[1];   // [L, 2^17, 2] f32
    f32x2*       out    = (f32x2*)d_out;           // [L, N, 2] f32 (concat of tuple)

    const int n = in_sizes[0] / 3;

    // Mirror the reference's numpy double-precision resolution computation
    // bit-for-bit (same libm exp/log/pow chain, then int() truncation).
    LevelRes lr;
    const double b = exp((log(256.0) - log(16.0)) / (double)(NUM_LEVELS - 1));
    for (int l = 0; l < NUM_LEVELS; ++l) {
        lr.r[l] = (float)(int)(16.0 * pow(b, (double)l));
    }

    if (n <= 0) return;
    const int grid = (n + BLOCK - 1) / BLOCK;
    MultiResHashEncoder_kernel<<<grid, BLOCK, 0, stream>>>(x, tables, out, n, lr);
}